// SoftRasLayer_69595650064583
// MI455X (gfx1250) — compile-verified
//
#include <hip/hip_runtime.h>
#include <math.h>

typedef __attribute__((ext_vector_type(2))) float v2f;
typedef __attribute__((ext_vector_type(8))) float v8f;
typedef __attribute__((ext_vector_type(4))) int   v4i;

#define NFACES    384
#define REC       36          // floats per face record (9 float4)
#define CROP0     16
#define CROPN     224
#define SIGMA_INV 1.0e5f
#define GAMMA_INV 1.0e5f
#define EPSV      1.0e-4f
#define NEARV     0.1f
#define FARV      100.0f
#define THRESH    1.1512915e-4f   // SIGMA * ln(1/DIST_EPS - 1)

#if defined(__has_builtin)
#if __has_builtin(__builtin_amdgcn_global_load_async_to_lds_b128)
#define HAVE_ASYNC_LDS 1
#endif
#if __has_builtin(__builtin_amdgcn_s_wait_asynccnt)
#define HAVE_WAIT_ASYNC_BUILTIN 1
#endif
#endif

// ---------------------------------------------------------------------------
// Kernel 1: per-face preprocessing.
// Record layout (floats):
//  [0..5]   a0,b0,c0,a1,b1,c1      (w0 = a0*qx+b0*qy+c0, w1 likewise)
//  [6..11]  ax,ay per edge k=0..2  (= x0,y0,x1,y1,x2,y2)
//  [12..17] ex,ey per edge
//  [18..20] 1/max(ex^2+ey^2,1e-12) per edge
//  [21..23] 1/z0, 1/z1, 1/z2
//  [24..32] tex[k][c] (k-major)
//  [33..35] pad
// ---------------------------------------------------------------------------
__global__ void prep_faces(const float* __restrict__ faces,
                           const float* __restrict__ tex,
                           float* __restrict__ tab) {
    int f = blockIdx.x * blockDim.x + threadIdx.x;
    if (f >= NFACES) return;
    const float* v = faces + f * 9;                   // (3 verts)x(x,y,z)
    float x0 = v[0], y0 = v[1], z0 = v[2];
    float x1 = v[3], y1 = v[4], z1 = v[5];
    float x2 = v[6], y2 = v[7], z2 = v[8];
    float den = (y1 - y2) * (x0 - x2) + (x2 - x1) * (y0 - y2);
    if (fabsf(den) < 1e-10f) den = (den < 0.0f) ? -1e-10f : 1e-10f;
    float a0 = (y1 - y2) / den, b0 = (x2 - x1) / den;
    float c0 = -(a0 * x2 + b0 * y2);
    float a1 = (y2 - y0) / den, b1 = (x0 - x2) / den;
    float c1 = -(a1 * x2 + b1 * y2);
    float* r = tab + f * REC;
    r[0] = a0; r[1] = b0; r[2] = c0; r[3] = a1; r[4] = b1; r[5] = c1;
    float vx[3] = {x0, x1, x2}, vy[3] = {y0, y1, y2};
#pragma unroll
    for (int k = 0; k < 3; ++k) {
        float ax = vx[k],           ay = vy[k];
        float bx = vx[(k + 1) % 3], by = vy[(k + 1) % 3];
        float ex = bx - ax, ey = by - ay;
        r[6 + 2 * k]  = ax; r[7 + 2 * k]  = ay;
        r[12 + 2 * k] = ex; r[13 + 2 * k] = ey;
        r[18 + k] = 1.0f / fmaxf(ex * ex + ey * ey, 1e-12f);
    }
    r[21] = 1.0f / z0; r[22] = 1.0f / z1; r[23] = 1.0f / z2;
    const float* t = tex + f * 9;
#pragma unroll
    for (int i = 0; i < 9; ++i) r[24 + i] = t[i];
    r[33] = 0.0f; r[34] = 0.0f; r[35] = 0.0f;
}

// ---------------------------------------------------------------------------
// Kernel 2: rasterize. One wave = 16 pixels (one row segment of the crop).
// WMMA F32 16x16x4: A = 16 faces x [a,b,c,0], B = [qx;qy;1;0] x 16 pixels,
// D = w for 16 faces x 16 pixels. Lane L owns pixel L%16; its 8 D-VGPRs hold
// faces r + 8*(L>=16). Each half-wave runs an online softmax over its half of
// the faces; halves merged at the end via __shfl_xor(...,16).
// Face table staged into LDS with GLOBAL_LOAD_ASYNC_TO_LDS_B128 (ASYNCcnt)
// when the builtin is available, else a plain float4 copy.
// Grid is exact (392 blocks x 256 thr = 3136 waves = 224 rows * 14 tiles), so
// EXEC is all-ones at every WMMA.
// ---------------------------------------------------------------------------
__global__ void __launch_bounds__(256) raster(const float* __restrict__ tab,
                                              float* __restrict__ seg,
                                              float* __restrict__ rgbout) {
    __shared__ float lds[NFACES * REC];              // 55296 B

    __builtin_prefetch(tab, 0, 3);                   // global_prefetch_b8
    {   // stage the face table into LDS (16B per lane per iteration)
        const int n4 = NFACES * REC / 4;             // 3456 x 16B chunks
#ifdef HAVE_ASYNC_LDS
        typedef __attribute__((address_space(1))) v4i AGv;   // global int4
        typedef __attribute__((address_space(3))) v4i ALv;   // LDS int4
        AGv* g = (AGv*)tab;
        ALv* l = (ALv*)lds;
        for (int i = threadIdx.x; i < n4; i += blockDim.x)
            __builtin_amdgcn_global_load_async_to_lds_b128(g + i, l + i, 0, 0);
#ifdef HAVE_WAIT_ASYNC_BUILTIN
        __builtin_amdgcn_s_wait_asynccnt(0);
#else
        asm volatile("s_wait_asynccnt 0" ::: "memory");
#endif
#else
        const float4* src = (const float4*)tab;
        float4* dst = (float4*)lds;
        for (int i = threadIdx.x; i < n4; i += blockDim.x) dst[i] = src[i];
#endif
    }
    __syncthreads();

    const int lane = threadIdx.x & 31;
    const int wave = (blockIdx.x * blockDim.x + threadIdx.x) >> 5;
    const int n    = lane & 15;                      // pixel column in tile (N)
    const int half = lane >> 4;                      // face half (M offset 0/8)
    const int row  = wave / (CROPN / 16);            // 0..223
    const int tcol = wave % (CROPN / 16);
    const int col  = tcol * 16 + n;                  // 0..223
    const int fr = row + CROP0, fc = col + CROP0;    // full-image coords
    const float qx = (2.0f * fc + 1.0f) / 256.0f - 1.0f;
    const float qy = (2.0f * (255 - fr) + 1.0f) / 256.0f - 1.0f;

    // B operand: VGPR0 = rows K0|K2 = qx | 1 ; VGPR1 = rows K1|K3 = qy | 0
    v2f B;
    B.x = half ? 1.0f : qx;
    B.y = half ? 0.0f : qy;

    // online-softmax state (per lane = per pixel, half of the faces)
    float m = EPSV, S = 0.0f, C0 = 0.0f, C1 = 0.0f, C2 = 0.0f, P = 1.0f;

    for (int tile = 0; tile < NFACES / 16; ++tile) {
        // A operand for face M = lane%16: lanes<16 carry K={0,1}=(a,b),
        // lanes>=16 carry K={2,3}=(c,0)
        const float* rA = lds + (tile * 16 + n) * REC;
        v2f A0, A1;
        if (half == 0) { A0.x = rA[0]; A0.y = rA[1]; A1.x = rA[3]; A1.y = rA[4]; }
        else           { A0.x = rA[2]; A0.y = 0.0f;  A1.x = rA[5]; A1.y = 0.0f; }
        v8f cz = {};
        v8f W0 = __builtin_amdgcn_wmma_f32_16x16x4_f32(false, A0, false, B,
                                                       (short)0, cz, false, false);
        v8f W1 = __builtin_amdgcn_wmma_f32_16x16x4_f32(false, A1, false, B,
                                                       (short)0, cz, false, false);
#pragma unroll
        for (int r = 0; r < 8; ++r) {
            const int f = tile * 16 + half * 8 + r;  // matches D-layout M index
            const float* rd = lds + f * REC;
            float w0 = W0[r], w1 = W1[r], w2 = 1.0f - w0 - w1;
            bool inside = (w0 >= 0.0f) & (w1 >= 0.0f) & (w2 >= 0.0f);

            // min squared distance to the 3 edge segments
            float d2min = 1e30f;
#pragma unroll
            for (int k = 0; k < 3; ++k) {
                float ax = rd[6 + 2 * k],  ay = rd[7 + 2 * k];
                float ex = rd[12 + 2 * k], ey = rd[13 + 2 * k];
                float il2 = rd[18 + k];
                float qax = qx - ax, qay = qy - ay;
                float t = fminf(fmaxf((qax * ex + qay * ey) * il2, 0.0f), 1.0f);
                float dx = qax - t * ex, dy = qay - t * ey;
                d2min = fminf(d2min, dx * dx + dy * dy);
            }
            float sgn = inside ? 1.0f : -1.0f;
            float Dp  = 1.0f / (1.0f + __expf(-sgn * d2min * SIGMA_INV));
            bool contrib = inside | (d2min <= THRESH);

            // clamped, renormalized barycentrics
            float u0 = fminf(fmaxf(w0, 0.0f), 1.0f);
            float u1 = fminf(fmaxf(w1, 0.0f), 1.0f);
            float u2 = fminf(fmaxf(w2, 0.0f), 1.0f);
            float inv = 1.0f / fmaxf(u0 + u1 + u2, 1e-12f);
            u0 *= inv; u1 *= inv; u2 *= inv;

            float zp = 1.0f / fmaxf(u0 * rd[21] + u1 * rd[22] + u2 * rd[23], 1e-12f);
            bool valid = contrib & (zp >= NEARV) & (zp <= FARV);
            float zn = (FARV - zp) / (FARV - NEARV);

            float col0 = u0 * rd[24] + u1 * rd[27] + u2 * rd[30];
            float col1 = u0 * rd[25] + u1 * rd[28] + u2 * rd[31];
            float col2 = u0 * rd[26] + u1 * rd[29] + u2 * rd[32];

            // online softmax update; invalid faces use zn=-1 so their exp
            // term underflows to exactly 0 (m >= EPSV always)
            float znv = valid ? zn : -1.0f;
            float mn  = fmaxf(m, znv);
            float sc  = __expf((m - mn) * GAMMA_INV);
            float e   = Dp * __expf((znv - mn) * GAMMA_INV);
            S  = S  * sc + e;
            C0 = C0 * sc + e * col0;
            C1 = C1 * sc + e * col1;
            C2 = C2 * sc + e * col2;
            m  = mn;
            P *= contrib ? (1.0f - Dp) : 1.0f;
        }
    }

    // merge the two half-wave softmax states (same pixel, disjoint faces)
    float mo  = __shfl_xor(m,  16, 32);
    float So  = __shfl_xor(S,  16, 32);
    float C0o = __shfl_xor(C0, 16, 32);
    float C1o = __shfl_xor(C1, 16, 32);
    float C2o = __shfl_xor(C2, 16, 32);
    float Po  = __shfl_xor(P,  16, 32);
    float mn = fmaxf(m, mo);
    float s1 = __expf((m  - mn) * GAMMA_INV);
    float s2 = __expf((mo - mn) * GAMMA_INV);
    S  = S  * s1 + So  * s2;
    C0 = C0 * s1 + C0o * s2;
    C1 = C1 * s1 + C1o * s2;
    C2 = C2 * s1 + C2o * s2;
    P *= Po;

    float bg   = __expf((EPSV - mn) * GAMMA_INV);
    float invd = 1.0f / (S + bg);
    if (half == 0) {
        int p = row * CROPN + col;
        seg[p]            = 1.0f - P;
        rgbout[p * 3 + 0] = C0 * invd;
        rgbout[p * 3 + 1] = C1 * invd;
        rgbout[p * 3 + 2] = C2 * invd;
    }
}

extern "C" void kernel_launch(void* const* d_in, const int* in_sizes, int n_in,
                              void* d_out, int out_size, void* d_ws, size_t ws_size,
                              hipStream_t stream) {
    const float* faces = (const float*)d_in[0];   // (1,384,3,3)
    const float* depth = (const float*)d_in[1];   // (1,384,3,3)
    float* tab = (float*)d_ws;                    // 384*36*4 = 55296 B scratch
    float* seg = (float*)d_out;                   // 224*224
    float* rgb = (float*)d_out + CROPN * CROPN;   // 224*224*3

    prep_faces<<<3, 128, 0, stream>>>(faces, depth, tab);
    // 224 rows * 14 tiles = 3136 waves; 8 waves/block -> 392 blocks exactly
    raster<<<392, 256, 0, stream>>>(tab, seg, rgb);
}